// LowRankMHA_35828617183991
// MI455X (gfx1250) — compile-verified
//
#include <hip/hip_runtime.h>
#include <stdint.h>

// ---------------------------------------------------------------------------
// LowRankMHA PGD solver for MI455X (gfx1250, wave32, WMMA).
// fp32 path via V_WMMA_F32_16X16X4_F32: ~70 GFLOP total, balanced against the
// ~2 GB of parameter streaming at 23.3 TB/s, so fp32 matrix cores are the
// right precision (bf16 would not buy wall-clock and hurts PGD fidelity).
// Workspace layout (floats): left(8.4M) right(8.4M) d_right(8.4M)
//                            sumsq(4096) R3(4096) G[4](1.05M)  ~= 105 MB.
// ---------------------------------------------------------------------------

typedef float v2f __attribute__((ext_vector_type(2)));
typedef float v8f __attribute__((ext_vector_type(8)));

#define PAIRS 64
#define SDIM 2048
#define RANK 64
#define DHEAD 64
#define NSTEPS 8
#define NPARAM (PAIRS * SDIM * RANK) /* 8388608 */
#define GBS 32                       /* s-block for gram/pv kernels */

// ---------------- threefry2x32-20 (JAX PRNG) -------------------------------
__host__ __device__ inline void tf2x32(uint32_t k0, uint32_t k1,
                                       uint32_t x0, uint32_t x1,
                                       uint32_t& y0, uint32_t& y1) {
  const uint32_t rotA[4] = {13, 15, 26, 6};
  const uint32_t rotB[4] = {17, 29, 16, 24};
  uint32_t ks[3] = {k0, k1, k0 ^ k1 ^ 0x1BD11BDAu};
  x0 += ks[0]; x1 += ks[1];
  for (int g = 0; g < 5; ++g) {
    const uint32_t* rot = (g & 1) ? rotB : rotA;
    for (int i = 0; i < 4; ++i) {
      x0 += x1;
      x1 = (x1 << rot[i]) | (x1 >> (32 - rot[i]));
      x1 ^= x0;
    }
    x0 += ks[(g + 1) % 3];
    x1 += ks[(g + 2) % 3] + (uint32_t)(g + 1);
  }
  y0 = x0; y1 = x1;
}

// ---------------- WMMA helpers (wave32 fragment layouts) -------------------
// A 16x4 fp32: lane l: m=l&15, k-pair base bk=(l>>4)*2 -> {A[m][k0+bk], A[m][k0+bk+1]}
// B 4x16 fp32: lane l: n=l&15, same k split across lane halves.
// C/D 16x16:   VGPR v: M = v + 8*(l>=16), N = l&15.

__device__ __forceinline__ v8f wmma4(v2f a, v2f b, v8f c) {
  return __builtin_amdgcn_wmma_f32_16x16x4_f32(false, a, false, b, (short)0, c,
                                               false, false);
}

// element-major (x index is slow axis): S[(x0+m)*ld + k0+bk]
__device__ __forceinline__ v2f fragA_xmaj(const float* S, int ld, int x0, int k0) {
  int l = threadIdx.x & 31, m = l & 15, bk = (l >> 4) << 1;
  const float* p = S + (x0 + m) * ld + k0 + bk;
  v2f a; a.x = p[0]; a.y = p[1]; return a;
}
// k-major (k index is slow axis): S[(k0+bk)*ld + x0+m]
__device__ __forceinline__ v2f frag_kmaj(const float* S, int ld, int k0, int x0) {
  int l = threadIdx.x & 31, m = l & 15, bk = (l >> 4) << 1;
  v2f a;
  a.x = S[(k0 + bk) * ld + x0 + m];
  a.y = S[(k0 + bk + 1) * ld + x0 + m];
  return a;
}
// A = (raw_row * ln[row])^2, x-major (left sphere squared on the fly)
__device__ __forceinline__ v2f fragA_xmaj_lsq(const float* S, int ld, int x0,
                                              int k0, const float* ln) {
  int l = threadIdx.x & 31, m = l & 15, bk = (l >> 4) << 1;
  float sc = ln[x0 + m];
  float a0 = S[(x0 + m) * ld + k0 + bk] * sc;
  float a1 = S[(x0 + m) * ld + k0 + bk + 1] * sc;
  v2f a; a.x = a0 * a0; a.y = a1 * a1; return a;
}
// B = (raw[k][x] * rin[k])^2, k-major (right sphere squared on the fly)
__device__ __forceinline__ v2f fragB_kmaj_rsq(const float* S, int ld, int k0,
                                              int x0, const float* rin) {
  int l = threadIdx.x & 31, m = l & 15, bk = (l >> 4) << 1;
  float b0 = S[(k0 + bk) * ld + x0 + m] * rin[k0 + bk];
  float b1 = S[(k0 + bk + 1) * ld + x0 + m] * rin[k0 + bk + 1];
  v2f b; b.x = b0 * b0; b.y = b1 * b1; return b;
}

__device__ __forceinline__ float safe_rinv(float ss) {
  return ss > 0.f ? rsqrtf(ss) : 0.f;
}

// ---------------- init: simplex init with JAX threefry ---------------------
__global__ __launch_bounds__(256) void init_params_kernel(
    float* lp, float* rp, uint32_t k1a, uint32_t k1b, uint32_t k2a, uint32_t k2b) {
  int idx = blockIdx.x * 256 + threadIdx.x;
  if (idx >= 2 * NPARAM) return;
  bool isL = idx < NPARAM;
  uint32_t e = (uint32_t)(isL ? idx : idx - NPARAM);
  uint32_t K0 = isL ? k1a : k2a, K1 = isL ? k1b : k2b;
  const uint32_t half = NPARAM / 2;
  uint32_t x0 = (e < half) ? e : e - half;
  uint32_t x1 = (e < half) ? e + half : e;
  uint32_t y0, y1;
  tf2x32(K0, K1, x0, x1, y0, y1);
  uint32_t bits = (e < half) ? y0 : y1;
  float u = __uint_as_float((bits >> 9) | 0x3f800000u) - 1.0f; // [0,1)
  float center = isL ? 0.125f : 0.022097086912079612f;         // 1/sqrt(64|2048)
  (isL ? lp : rp)[e] = center + 0.002f * u - 0.001f;
}

__global__ __launch_bounds__(256) void zero_kernel(float* p, int n) {
  int i = blockIdx.x * 256 + threadIdx.x;
  if (i < n) p[i] = 0.f;
}

// ---------------- K1: sum of squares over s per right row ------------------
__global__ __launch_bounds__(256) void right_norms_kernel(const float* rp,
                                                          float* ssb) {
  int pair = blockIdx.x >> 2, chunk = blockIdx.x & 3;
  int t = threadIdx.x;
  __shared__ float acc[RANK];
  if (t < RANK) acc[t] = 0.f;
  __syncthreads();
  int r = t >> 2, so = t & 3;
  const float* row = rp + ((pair * RANK) + r) * SDIM + chunk * 512;
  float s = 0.f;
  for (int i = 0; i < 128; ++i) {
    float v = row[so + 4 * i];
    s += v * v;
  }
  atomicAdd(&acc[r], s);
  __syncthreads();
  if (t < RANK) atomicAdd(ssb + pair * RANK + t, acc[t]);
}

// ---------------- K2: four 64x64 Grams via WMMA, reduce over s -------------
// g=0: G_RR = Rsq @ RsqT   g=1: G_KR = kT @ RsqT
// g=2: G_LL = LsqT @ Lsq   g=3: G_LQ = LsqT @ q  (q pre-scaled by 1/8)
__global__ __launch_bounds__(256) void gram_kernel(const float* lp, const float* rp,
                                                   const float* qg, const float* kg,
                                                   const float* ssb, float* G) {
  __shared__ float sRsq[RANK * (GBS + 1)]; // [r][s_local]
  __shared__ float sL[GBS * 65];           // [s_local][r] raw -> sphere^2
  __shared__ float sQ[GBS * 65];           // [s_local][d] * 0.125
  __shared__ float sK[GBS * 65];           // [s_local][d]
  __shared__ float rinv[RANK];
  __shared__ float lnrm[GBS];
  int pair = blockIdx.x >> 3, chunk = blockIdx.x & 7;
  int t = threadIdx.x, wave = t >> 5;
  if (t < RANK) rinv[t] = safe_rinv(ssb[pair * RANK + t]);
  v8f acc[8];
#pragma unroll
  for (int i = 0; i < 8; ++i) acc[i] = {0.f, 0.f, 0.f, 0.f, 0.f, 0.f, 0.f, 0.f};
  int g = wave >> 1, halfw = wave & 1;
  for (int blk = 0; blk < 8; ++blk) {
    int s0 = chunk * 256 + blk * GBS;
    __syncthreads();
#pragma unroll
    for (int i = 0; i < 8; ++i) {
      int e = t + 256 * i;
      { int rr = e >> 5, sl = e & 31;
        float x = rp[((pair * RANK) + rr) * SDIM + s0 + sl] * rinv[rr];
        sRsq[rr * (GBS + 1) + sl] = x * x; }
      { int sr = e >> 6, c = e & 63;
        int gi = ((pair * SDIM) + s0 + sr) * DHEAD + c;
        sL[sr * 65 + c] = lp[gi];
        sQ[sr * 65 + c] = qg[gi] * 0.125f;
        sK[sr * 65 + c] = kg[gi]; }
    }
    __syncthreads();
    if (t < GBS) {
      float sm = 0.f;
      const float* lr = &sL[t * 65];
      for (int rr = 0; rr < RANK; ++rr) { float x = lr[rr]; sm += x * x; }
      lnrm[t] = safe_rinv(sm);
    }
    __syncthreads();
#pragma unroll
    for (int i = 0; i < 8; ++i) { // convert sL to sphere^2
      int e = t + 256 * i; int sr = e >> 6, c = e & 63;
      float x = sL[sr * 65 + c] * lnrm[sr];
      sL[sr * 65 + c] = x * x;
    }
    __syncthreads();
#pragma unroll
    for (int tt = 0; tt < 8; ++tt) {
      int idx = halfw * 8 + tt, ti = (idx >> 2) * 16, tj = (idx & 3) * 16;
      v8f c = acc[tt];
#pragma unroll
      for (int kk = 0; kk < GBS; kk += 4) {
        v2f a, b;
        if (g == 0)      { a = fragA_xmaj(sRsq, GBS + 1, ti, kk); b = fragA_xmaj(sRsq, GBS + 1, tj, kk); }
        else if (g == 1) { a = frag_kmaj(sK, 65, kk, ti);         b = fragA_xmaj(sRsq, GBS + 1, tj, kk); }
        else if (g == 2) { a = frag_kmaj(sL, 65, kk, ti);         b = frag_kmaj(sL, 65, kk, tj); }
        else             { a = frag_kmaj(sL, 65, kk, ti);         b = frag_kmaj(sQ, 65, kk, tj); }
        c = wmma4(a, b, c);
      }
      acc[tt] = c;
    }
  }
  float* Gp = G + ((g * PAIRS) + pair) * 4096;
  int l = t & 31, n = l & 15, hi = (l >> 4) * 8;
  for (int tt = 0; tt < 8; ++tt) {
    int idx = halfw * 8 + tt, ti = (idx >> 2) * 16, tj = (idx & 3) * 16;
#pragma unroll
    for (int v = 0; v < 8; ++v)
      atomicAdd(Gp + (ti + v + hi) * 64 + tj + n, acc[tt][v]);
  }
}

// ---------------- K3: left update (fused) + d_right + R3 partials ----------
__global__ __launch_bounds__(256) void update_kernel(float* lp, const float* rp,
                                                     const float* qg, const float* kg,
                                                     const float* ssb, const float* G,
                                                     float* r3, float* dr) {
  __shared__ float sA[64 * 65]; // left raw rows | right raw rows
  __shared__ float sB[64 * 65]; // q rows (then D) | k rows (then D2)
  __shared__ float lnr[64];
  __shared__ float tsum[64];
  int pair = blockIdx.x >> 5, chunk = blockIdx.x & 31;
  int s0 = chunk * 64;
  int t = threadIdx.x, wave = t >> 5;
  int l = t & 31, n = l & 15, hi = (l >> 4) * 8;
  const float* GRR = G + (0 * PAIRS + pair) * 4096;
  const float* GKR = G + (1 * PAIRS + pair) * 4096;
  const float* GLL = G + (2 * PAIRS + pair) * 4096;
  const float* GLQ = G + (3 * PAIRS + pair) * 4096;
  const v8f vz = {0.f, 0.f, 0.f, 0.f, 0.f, 0.f, 0.f, 0.f};
  v8f dacc[2];

  // ---------- left phase: d_left = Lsq@G_RR - q@G_KR ----------
#pragma unroll
  for (int i = 0; i < 16; ++i) {
    int e = t + 256 * i; int sr = e >> 6, c = e & 63;
    int gi = ((pair * SDIM) + s0 + sr) * DHEAD + c;
    sA[sr * 65 + c] = lp[gi];
    sB[sr * 65 + c] = qg[gi] * 0.125f;
  }
  __syncthreads();
  if (t < 64) {
    float sm = 0.f;
    for (int rr = 0; rr < RANK; ++rr) { float x = sA[t * 65 + rr]; sm += x * x; }
    lnr[t] = safe_rinv(sm);
  }
  __syncthreads();
#pragma unroll
  for (int tt = 0; tt < 2; ++tt) {
    int idx = wave * 2 + tt, ti = (idx >> 2) * 16, tj = (idx & 3) * 16;
    v8f c1 = vz, c2 = vz;
#pragma unroll
    for (int kk = 0; kk < 64; kk += 4) {
      c1 = wmma4(fragA_xmaj_lsq(sA, 65, ti, kk, lnr), frag_kmaj(GRR, 64, kk, tj), c1);
      c2 = wmma4(fragA_xmaj(sB, 65, ti, kk),          frag_kmaj(GKR, 64, kk, tj), c2);
    }
    dacc[tt] = c1 - c2;
  }
  __syncthreads(); // everyone done reading q tile -> reuse sB for D
#pragma unroll
  for (int tt = 0; tt < 2; ++tt) {
    int idx = wave * 2 + tt, ti = (idx >> 2) * 16, tj = (idx & 3) * 16;
#pragma unroll
    for (int v = 0; v < 8; ++v) sB[(ti + v + hi) * 65 + tj + n] = dacc[tt][v];
  }
  __syncthreads();
  if (t < 64) { // row projection coefficient
    float ln = lnr[t]; float sm = 0.f;
    for (int rr = 0; rr < RANK; ++rr) {
      float sph = sA[t * 65 + rr] * ln;
      float d1 = sB[t * 65 + rr] * 2.f * sph;
      sm += sph * d1;
    }
    tsum[t] = sm;
  }
  __syncthreads();
#pragma unroll
  for (int i = 0; i < 16; ++i) {
    int e = t + 256 * i; int sr = e >> 6, c = e & 63;
    float ln = lnr[sr], raw = sA[sr * 65 + c], sph = raw * ln;
    float d1 = sB[sr * 65 + c] * 2.f * sph;
    lp[((pair * SDIM) + s0 + sr) * DHEAD + c] = raw - (d1 - sph * tsum[sr]) * ln;
  }
  __syncthreads();

  // ---------- right phase: d_right = G_LL@Rsq - G_LQ@kT ----------
#pragma unroll
  for (int i = 0; i < 16; ++i) {
    int e = t + 256 * i;
    { int rr = e >> 6, sl = e & 63;
      sA[rr * 65 + sl] = rp[((pair * RANK) + rr) * SDIM + s0 + sl]; }
    { int sr = e >> 6, c = e & 63;
      sB[sr * 65 + c] = kg[((pair * SDIM) + s0 + sr) * DHEAD + c]; }
  }
  if (t < 64) lnr[t] = safe_rinv(ssb[pair * RANK + t]); // reuse lnr as rinv
  __syncthreads();
#pragma unroll
  for (int tt = 0; tt < 2; ++tt) {
    int idx = wave * 2 + tt, ti = (idx >> 2) * 16, tj = (idx & 3) * 16;
    v8f c1 = vz, c2 = vz;
#pragma unroll
    for (int kk = 0; kk < 64; kk += 4) {
      c1 = wmma4(fragA_xmaj(GLL, 64, ti, kk), fragB_kmaj_rsq(sA, 65, kk, tj, lnr), c1);
      c2 = wmma4(fragA_xmaj(GLQ, 64, ti, kk), fragA_xmaj(sB, 65, tj, kk),          c2);
    }
    dacc[tt] = c1 - c2;
  }
  __syncthreads(); // done reading k tile -> reuse sB for D2 [r][s]
#pragma unroll
  for (int tt = 0; tt < 2; ++tt) {
    int idx = wave * 2 + tt, ti = (idx >> 2) * 16, tj = (idx & 3) * 16;
#pragma unroll
    for (int v = 0; v < 8; ++v) sB[(ti + v + hi) * 65 + tj + n] = dacc[tt][v];
  }
  __syncthreads();
  if (t < 64) { // partial R3[r] = sum_s sphere * d1
    float in = lnr[t]; float sm = 0.f;
    for (int sl = 0; sl < 64; ++sl) {
      float sph = sA[t * 65 + sl] * in;
      float d1 = sB[t * 65 + sl] * 2.f * sph;
      sm += sph * d1;
    }
    atomicAdd(&r3[pair * RANK + t], sm);
  }
#pragma unroll
  for (int i = 0; i < 16; ++i) {
    int e = t + 256 * i; int rr = e >> 6, sl = e & 63;
    float in = lnr[rr], sph = sA[rr * 65 + sl] * in;
    float d1 = sB[rr * 65 + sl] * 2.f * sph;
    dr[((pair * RANK) + rr) * SDIM + s0 + sl] = d1;
  }
}

// ---------------- K4: finalize right update --------------------------------
__global__ __launch_bounds__(256) void finalize_right_kernel(float* rp, const float* dr,
                                                             const float* ssb,
                                                             const float* r3) {
  int pair = blockIdx.x >> 5, chunk = blockIdx.x & 31;
  int s0 = chunk * 64, t = threadIdx.x;
  __shared__ float rin[64], rr3[64];
  if (t < 64) {
    rin[t] = safe_rinv(ssb[pair * RANK + t]);
    rr3[t] = r3[pair * RANK + t];
  }
  __syncthreads();
#pragma unroll
  for (int i = 0; i < 16; ++i) {
    int e = t + 256 * i; int r_ = e >> 6, sl = e & 63;
    int gi = ((pair * RANK) + r_) * SDIM + s0 + sl;
    float raw = rp[gi], in = rin[r_], sph = raw * in;
    rp[gi] = raw - (dr[gi] - sph * rr3[r_]) * in;
  }
}

// ---------------- K5: P = Rsq @ value (64x64 per pair) ---------------------
__global__ __launch_bounds__(256) void pv_kernel(const float* rp, const float* vg,
                                                 const float* ssb, float* G) {
  __shared__ float sRsq[RANK * (GBS + 1)];
  __shared__ float sV[GBS * 65];
  __shared__ float rinv[RANK];
  int pair = blockIdx.x >> 3, chunk = blockIdx.x & 7;
  int t = threadIdx.x, wave = t >> 5;
  if (t < RANK) rinv[t] = safe_rinv(ssb[pair * RANK + t]);
  v8f acc[2];
  acc[0] = {0.f, 0.f, 0.f, 0.f, 0.f, 0.f, 0.f, 0.f};
  acc[1] = acc[0];
  for (int blk = 0; blk < 8; ++blk) {
    int s0 = chunk * 256 + blk * GBS;
    __syncthreads();
#pragma unroll
    for (int i = 0; i < 8; ++i) {
      int e = t + 256 * i;
      { int rr = e >> 5, sl = e & 31;
        float x = rp[((pair * RANK) + rr) * SDIM + s0 + sl] * rinv[rr];
        sRsq[rr * (GBS + 1) + sl] = x * x; }
      { int sr = e >> 6, c = e & 63;
        sV[sr * 65 + c] = vg[((pair * SDIM) + s0 + sr) * DHEAD + c]; }
    }
    __syncthreads();
#pragma unroll
    for (int tt = 0; tt < 2; ++tt) {
      int idx = wave * 2 + tt, ti = (idx >> 2) * 16, tj = (idx & 3) * 16;
      v8f c = acc[tt];
#pragma unroll
      for (int kk = 0; kk < GBS; kk += 4)
        c = wmma4(fragA_xmaj(sRsq, GBS + 1, ti, kk), frag_kmaj(sV, 65, kk, tj), c);
      acc[tt] = c;
    }
  }
  float* P = G + pair * 4096;
  int l = t & 31, n = l & 15, hi = (l >> 4) * 8;
  for (int tt = 0; tt < 2; ++tt) {
    int idx = wave * 2 + tt, ti = (idx >> 2) * 16, tj = (idx & 3) * 16;
#pragma unroll
    for (int v = 0; v < 8; ++v)
      atomicAdd(P + (ti + v + hi) * 64 + tj + n, acc[tt][v]);
  }
}

// ---------------- K6: out = Lsq @ P ----------------------------------------
__global__ __launch_bounds__(256) void out_kernel(const float* lp, const float* G,
                                                  float* out) {
  __shared__ float sL[64 * 65];
  __shared__ float lnr[64];
  int pair = blockIdx.x >> 5, chunk = blockIdx.x & 31;
  int s0 = chunk * 64, t = threadIdx.x, wave = t >> 5;
#pragma unroll
  for (int i = 0; i < 16; ++i) {
    int e = t + 256 * i; int sr = e >> 6, c = e & 63;
    sL[sr * 65 + c] = lp[((pair * SDIM) + s0 + sr) * DHEAD + c];
  }
  __syncthreads();
  if (t < 64) {
    float sm = 0.f;
    for (int rr = 0; rr < RANK; ++rr) { float x = sL[t * 65 + rr]; sm += x * x; }
    lnr[t] = safe_rinv(sm);
  }
  __syncthreads();
  const float* P = G + pair * 4096;
  int l = t & 31, n = l & 15, hi = (l >> 4) * 8;
#pragma unroll
  for (int tt = 0; tt < 2; ++tt) {
    int idx = wave * 2 + tt, ti = (idx >> 2) * 16, tj = (idx & 3) * 16;
    v8f c = {0.f, 0.f, 0.f, 0.f, 0.f, 0.f, 0.f, 0.f};
#pragma unroll
    for (int kk = 0; kk < 64; kk += 4)
      c = wmma4(fragA_xmaj_lsq(sL, 65, ti, kk, lnr), frag_kmaj(P, 64, kk, tj), c);
#pragma unroll
    for (int v = 0; v < 8; ++v)
      out[((pair * SDIM) + s0 + ti + v + hi) * DHEAD + tj + n] = c[v];
  }
}

// ---------------- driver ---------------------------------------------------
extern "C" void kernel_launch(void* const* d_in, const int* in_sizes, int n_in,
                              void* d_out, int out_size, void* d_ws, size_t ws_size,
                              hipStream_t stream) {
  (void)in_sizes; (void)n_in; (void)out_size; (void)ws_size;
  const float* qg = (const float*)d_in[0];
  const float* kg = (const float*)d_in[1];
  const float* vg = (const float*)d_in[2];
  float* out = (float*)d_out;
  float* ws = (float*)d_ws;
  float* lp = ws;                 // [pair][s][r]
  float* rp = lp + NPARAM;        // [pair][r][s]
  float* dr = rp + NPARAM;        // [pair][r][s]
  float* ssb = dr + NPARAM;       // [pair][r] right sumsq
  float* r3 = ssb + PAIRS * RANK; // [pair][r] projection sums
  float* G = r3 + PAIRS * RANK;   // [4][pair][64][64] (P reuses g=0 slot)
  const int ZN = 2 * PAIRS * RANK + 4 * PAIRS * 64 * 64; // accumulators to zero

  // JAX: k1,k2 = split(key(42)) via threefry on counts iota(4)
  uint32_t a0, a1, b0, b1;
  tf2x32(0u, 42u, 0u, 2u, a0, b0);
  tf2x32(0u, 42u, 1u, 3u, a1, b1);
  // key1 = (a0, a1), key2 = (b0, b1)

  init_params_kernel<<<(2 * NPARAM + 255) / 256, 256, 0, stream>>>(lp, rp, a0, a1, b0, b1);
  for (int step = 0; step < NSTEPS; ++step) {
    zero_kernel<<<(ZN + 255) / 256, 256, 0, stream>>>(ssb, ZN);
    right_norms_kernel<<<PAIRS * 4, 256, 0, stream>>>(rp, ssb);
    gram_kernel<<<PAIRS * 8, 256, 0, stream>>>(lp, rp, qg, kg, ssb, G);
    update_kernel<<<PAIRS * 32, 256, 0, stream>>>(lp, rp, qg, kg, ssb, G, r3, dr);
    finalize_right_kernel<<<PAIRS * 32, 256, 0, stream>>>(rp, dr, ssb, r3);
  }
  zero_kernel<<<(ZN + 255) / 256, 256, 0, stream>>>(ssb, ZN);
  right_norms_kernel<<<PAIRS * 4, 256, 0, stream>>>(rp, ssb);
  pv_kernel<<<PAIRS * 8, 256, 0, stream>>>(rp, vg, ssb, G);
  out_kernel<<<PAIRS * 32, 256, 0, stream>>>(lp, G, out);
}